// Linear_59502476919112
// MI455X (gfx1250) — compile-verified
//
#include <hip/hip_runtime.h>
#include <hip/hip_bf16.h>

// y[n,d,b] = w[idx[n],d] * x[n,d,b]   (N=16384, U=1024, D=128, B=16, fp32)
// Pure streaming kernel, ~270 MB of traffic for 33.5 MFLOP -> HBM-bound
// (0.12 FLOP/byte; floor ~11.6 us at 23.3 TB/s). WMMA is irrelevant here;
// the MI455X feature that matters is the async global->LDS DMA engine
// (ASYNCcnt path) for deep memory-level parallelism at zero VGPR cost.
//
// Per wave: double-buffered async DMA of whole 8 KB x rows + the gathered
// 512 B weight row, then LDS->VGPR b128 reads, VALU multiply, NT b128 stores.
// x loads are th:TH_LOAD_NT (read-once stream must not evict the hot 512 KB
// weight table from L2); weight-row loads stay RT (reused ~16x across waves).

#define WPB 2            // waves per block
#define TPB (WPB * 32)   // 64 threads
#define D_DIM 128
#define B_DIM 16
#define ROW_F (D_DIM * B_DIM)   // 2048 floats per row
#define ROW_BYTES (ROW_F * 4)   // 8192 bytes per row

typedef __attribute__((ext_vector_type(4))) float v4f;

// Issue one full stage: 16 async b128 loads for the 8 KB x row + 1 async b128
// load for the 512 B gathered weight row. Each lane moves 16 B per
// instruction; the instruction offset advances BOTH the global and the LDS
// address (CDNA5 ISA GVS semantics), so one LDS-addr VGPR and one
// voffset VGPR cover all 16 chunks.
__device__ __forceinline__ void async_stage(unsigned lds_x, unsigned lds_w,
                                            unsigned voff,
                                            const float* xrow,
                                            const float* wrow) {
  asm volatile(
      "global_load_async_to_lds_b128 %0, %1, %2 offset:0 th:TH_LOAD_NT\n\t"
      "global_load_async_to_lds_b128 %0, %1, %2 offset:512 th:TH_LOAD_NT\n\t"
      "global_load_async_to_lds_b128 %0, %1, %2 offset:1024 th:TH_LOAD_NT\n\t"
      "global_load_async_to_lds_b128 %0, %1, %2 offset:1536 th:TH_LOAD_NT\n\t"
      "global_load_async_to_lds_b128 %0, %1, %2 offset:2048 th:TH_LOAD_NT\n\t"
      "global_load_async_to_lds_b128 %0, %1, %2 offset:2560 th:TH_LOAD_NT\n\t"
      "global_load_async_to_lds_b128 %0, %1, %2 offset:3072 th:TH_LOAD_NT\n\t"
      "global_load_async_to_lds_b128 %0, %1, %2 offset:3584 th:TH_LOAD_NT\n\t"
      "global_load_async_to_lds_b128 %0, %1, %2 offset:4096 th:TH_LOAD_NT\n\t"
      "global_load_async_to_lds_b128 %0, %1, %2 offset:4608 th:TH_LOAD_NT\n\t"
      "global_load_async_to_lds_b128 %0, %1, %2 offset:5120 th:TH_LOAD_NT\n\t"
      "global_load_async_to_lds_b128 %0, %1, %2 offset:5632 th:TH_LOAD_NT\n\t"
      "global_load_async_to_lds_b128 %0, %1, %2 offset:6144 th:TH_LOAD_NT\n\t"
      "global_load_async_to_lds_b128 %0, %1, %2 offset:6656 th:TH_LOAD_NT\n\t"
      "global_load_async_to_lds_b128 %0, %1, %2 offset:7168 th:TH_LOAD_NT\n\t"
      "global_load_async_to_lds_b128 %0, %1, %2 offset:7680 th:TH_LOAD_NT\n\t"
      "global_load_async_to_lds_b128 %3, %1, %4 offset:0"
      :
      : "v"(lds_x), "v"(voff), "s"(xrow), "v"(lds_w), "s"(wrow)
      : "memory");
}

__device__ __forceinline__ void wait_async_le17() {
  asm volatile("s_wait_asynccnt 17" ::: "memory");
}
__device__ __forceinline__ void wait_async_0() {
  asm volatile("s_wait_asynccnt 0" ::: "memory");
}

__global__ void __launch_bounds__(TPB) diag_gather_mul_kernel(
    const float* __restrict__ x,     // [N, D, B]
    const float* __restrict__ w,     // [U, D]
    const int* __restrict__ idx,     // [N]
    float* __restrict__ out,         // [N, D, B]
    int N) {
  __shared__ __align__(16) float sx[WPB][2][ROW_F];  // 2 x 8 KB per wave
  __shared__ __align__(16) float sw[WPB][2][D_DIM];  // 2 x 512 B per wave

  const int lane = threadIdx.x & 31;
  const int wv = threadIdx.x >> 5;
  const int wave_id = blockIdx.x * WPB + wv;
  const int stride = gridDim.x * WPB;

  const unsigned voff = (unsigned)(lane * 16);
  // Low 32 bits of a flat pointer into LDS == wave-relative LDS byte address
  // (ISA aperture rule).
  const unsigned sx0 = (unsigned)(size_t)(&sx[wv][0][0]) + voff;
  const unsigned sx1 = (unsigned)(size_t)(&sx[wv][1][0]) + voff;
  const unsigned sw0 = (unsigned)(size_t)(&sw[wv][0][0]) + voff;
  const unsigned sw1 = (unsigned)(size_t)(&sw[wv][1][0]) + voff;

  int n = wave_id;
  if (n >= N) return;

  {
    const int id0 = idx[n];
    async_stage(sx0, sw0, voff, x + (size_t)n * ROW_F,
                w + (size_t)id0 * D_DIM);
  }

  int buf = 0;
  while (true) {
    const int n_next = n + stride;
    const bool has_next = (n_next < N);
    if (has_next) {
      const int id1 = idx[n_next];
      async_stage(buf ? sx0 : sx1, buf ? sw0 : sw1, voff,
                  x + (size_t)n_next * ROW_F, w + (size_t)id1 * D_DIM);
      // 17 ops just issued for stage k+1; async loads retire in order, so
      // ASYNCcnt <= 17 guarantees stage k's 17 transfers have landed in LDS.
      wait_async_le17();
    } else {
      wait_async_0();
    }

    const float* xr = &sx[wv][buf][0];
    const float* wr = &sw[wv][buf][0];
    float* outrow = out + (size_t)n * ROW_F;

#pragma unroll
    for (int i = 0; i < 16; ++i) {
      const int c = i * 32 + lane;         // float4 chunk within the row
      const int d = i * 8 + (lane >> 2);   // d index of this chunk (B=16)
      const float wt = wr[d];              // ds_load_b32, broadcast in 4-lane groups
      v4f v = *(const v4f*)(xr + c * 4);   // ds_load_b128
      v4f r = v * wt;
      // NT store: 128 MB output stream must not evict the weight table.
      __builtin_nontemporal_store(r, (v4f*)(outrow + (size_t)c * 4));
    }

    if (!has_next) break;
    n = n_next;
    buf ^= 1;
  }
}

extern "C" void kernel_launch(void* const* d_in, const int* in_sizes, int n_in,
                              void* d_out, int out_size, void* d_ws,
                              size_t ws_size, hipStream_t stream) {
  const float* x = (const float*)d_in[0];   // [N, D, B] fp32
  const float* w = (const float*)d_in[1];   // [U, D]    fp32
  const int* idx = (const int*)d_in[2];     // [N]       int
  float* out = (float*)d_out;

  const int N = in_sizes[2];

  // 2048 blocks x 2 waves = 4096 waves -> 4 rows per wave, ~35 MB of async
  // traffic in flight in steady state (>= BW * latency for 23.3 TB/s HBM).
  int blocks = 2048;
  if (blocks * WPB > N) blocks = (N + WPB - 1) / WPB;

  diag_gather_mul_kernel<<<blocks, TPB, 0, stream>>>(x, w, idx, out, N);
}